// GNN_32873679683699
// MI455X (gfx1250) — compile-verified
//
#include <hip/hip_runtime.h>
#include <hip/hip_bf16.h>

#define N_NODES    100000
#define N_EDGES    1000000
#define CH         64
#define NUM_GRAPHS 1024

typedef __attribute__((ext_vector_type(2))) float v2f;
typedef __attribute__((ext_vector_type(8))) float v8f;

// ---------------- degree / normalization ----------------
__global__ void k_deg(const int* __restrict__ dst, float* __restrict__ deg) {
    int e = blockIdx.x * blockDim.x + threadIdx.x;
    if (e < N_EDGES) atomicAdd(&deg[dst[e]], 1.0f);
}

// in-place: deg -> rsqrt(deg + 1)  (self loop included)
__global__ void k_dinv(float* __restrict__ deg) {
    int i = blockIdx.x * blockDim.x + threadIdx.x;
    if (i < N_NODES) deg[i] = rsqrtf(deg[i] + 1.0f);
}

// ---------------- Y[N,64] = X[N,64] @ W[64,64] via V_WMMA_F32_16X16X4_F32 ----
// 256 threads = 8 waves; each wave computes one 16-row tile across all 4
// 16-col tiles (4 live v8f accumulators -> A fragment reused 4x per K-step).
// W staged in LDS once per block.
__global__ void __launch_bounds__(256) k_gemm64(const float* __restrict__ X,
                                                const float* __restrict__ W,
                                                float* __restrict__ Y,
                                                int nrows) {
    __shared__ float sW[64 * 64];
    for (int i = threadIdx.x; i < 64 * 64; i += 256) sW[i] = W[i];
    __syncthreads();

    const int wave = threadIdx.x >> 5;       // 0..7
    const int lane = threadIdx.x & 31;
    const int tile = blockIdx.x * 8 + wave;  // 16-row tile index
    if (tile * 16 >= nrows) return;          // wave-uniform: EXEC stays all-1s
    const int row0  = tile * 16;
    const int m     = lane & 15;             // A row / B,C col within tile
    const int khalf = lane >> 4;             // 0: K={0,1}  1: K={2,3}

    v8f acc0 = {}, acc1 = {}, acc2 = {}, acc3 = {};
#pragma unroll
    for (int k0 = 0; k0 < 64; k0 += 4) {
        // A 16x4 fragment (ISA 7.12.2, 32-bit A): lane holds K = khalf*2 + {0,1}
        const float* ap = X + (size_t)(row0 + m) * CH + k0 + khalf * 2;
        v2f a; a.x = ap[0]; a.y = ap[1];
        const int kb = k0 + khalf * 2;
        v2f b0, b1, b2, b3;
        b0.x = sW[(kb + 0) * 64 +  0 + m]; b0.y = sW[(kb + 1) * 64 +  0 + m];
        b1.x = sW[(kb + 0) * 64 + 16 + m]; b1.y = sW[(kb + 1) * 64 + 16 + m];
        b2.x = sW[(kb + 0) * 64 + 32 + m]; b2.y = sW[(kb + 1) * 64 + 32 + m];
        b3.x = sW[(kb + 0) * 64 + 48 + m]; b3.y = sW[(kb + 1) * 64 + 48 + m];
        acc0 = __builtin_amdgcn_wmma_f32_16x16x4_f32(false, a, false, b0, (short)0, acc0, false, false);
        acc1 = __builtin_amdgcn_wmma_f32_16x16x4_f32(false, a, false, b1, (short)0, acc1, false, false);
        acc2 = __builtin_amdgcn_wmma_f32_16x16x4_f32(false, a, false, b2, (short)0, acc2, false, false);
        acc3 = __builtin_amdgcn_wmma_f32_16x16x4_f32(false, a, false, b3, (short)0, acc3, false, false);
    }
    // D layout: VGPR r, lanes<16 -> M=r ; lanes>=16 -> M=r+8 ; N = lane&15
    const int mrow = row0 + 8 * khalf;
#pragma unroll
    for (int r = 0; r < 8; ++r) {
        float* yp = Y + (size_t)(mrow + r) * CH + m;
        yp[0]  = acc0[r];
        yp[16] = acc1[r];
        yp[32] = acc2[r];
        yp[48] = acc3[r];
    }
}

// ---------------- per-edge message scatter (one wave per edge) --------------
// agg[dst] += xw[src] * dinv[src]*dinv[dst]; 64 ch -> float2 per lane.
// Entire xw/agg working set (51 MB) is L2-resident on MI455X (192 MB L2),
// so the f32 atomics resolve in the L2 atomic units.
__global__ void __launch_bounds__(256) k_scatter(const int* __restrict__ src,
                                                 const int* __restrict__ dst,
                                                 const float* __restrict__ dinv,
                                                 const float* __restrict__ xw,
                                                 float* __restrict__ agg) {
    int t = blockIdx.x * blockDim.x + threadIdx.x;
    int e = t >> 5;
    int lane = t & 31;
    if (e >= N_EDGES) return;
    int s = src[e], d = dst[e];
    float norm = dinv[s] * dinv[d];
    float2 v = ((const float2*)(xw + (size_t)s * CH))[lane];
    float* ap = agg + (size_t)d * CH + lane * 2;
    atomicAdd(ap + 0, v.x * norm);
    atomicAdd(ap + 1, v.y * norm);
}

// ---------------- self-loop + bias + ReLU ----------------
__global__ void k_combine(const float* __restrict__ agg, const float* __restrict__ xw,
                          const float* __restrict__ dinv, const float* __restrict__ bias,
                          float* __restrict__ h) {
    int t = blockIdx.x * blockDim.x + threadIdx.x;
    if (t >= N_NODES * CH) return;
    int node = t >> 6, c = t & 63;
    float di = dinv[node];                       // self-loop norm = 1/deg = dinv^2
    float v = agg[t] + xw[t] * (di * di) + bias[c];
    h[t] = fmaxf(v, 0.0f);
}

// ---------------- mean-pool accumulate (one wave per node) ----------------
__global__ void __launch_bounds__(256) k_pool(const int* __restrict__ batch,
                                              const float* __restrict__ h,
                                              float* __restrict__ sums,
                                              float* __restrict__ cnt) {
    int t = blockIdx.x * blockDim.x + threadIdx.x;
    int node = t >> 5;
    int lane = t & 31;
    if (node >= N_NODES) return;
    int g = batch[node];
    float2 v = ((const float2*)(h + (size_t)node * CH))[lane];
    float* sp = sums + (size_t)g * CH + lane * 2;
    atomicAdd(sp + 0, v.x);
    atomicAdd(sp + 1, v.y);
    if (lane == 0) atomicAdd(&cnt[g], 1.0f);
}

// ---------------- output GEMV: one wave per graph ----------------
__global__ void k_out(const float* __restrict__ sums, const float* __restrict__ cnt,
                      const float* __restrict__ Wout, const float* __restrict__ bout,
                      float* __restrict__ out) {
    int t = blockIdx.x * blockDim.x + threadIdx.x;
    int g = t >> 5;
    int lane = t & 31;
    if (g >= NUM_GRAPHS) return;
    float c = fmaxf(cnt[g], 1.0f);
    float2 v = ((const float2*)(sums + (size_t)g * CH))[lane];
    float2 w = ((const float2*)Wout)[lane];
    float partial = (v.x * w.x + v.y * w.y) / c;
#pragma unroll
    for (int off = 16; off > 0; off >>= 1)
        partial += __shfl_down(partial, off, 32);
    if (lane == 0) out[g] = partial + bout[0];
}

extern "C" void kernel_launch(void* const* d_in, const int* in_sizes, int n_in,
                              void* d_out, int out_size, void* d_ws, size_t ws_size,
                              hipStream_t stream) {
    const float* x    = (const float*)d_in[0];
    const int*   edge = (const int*)d_in[1];   // [2, E]: row 0 = src, row 1 = dst
    const int*   src  = edge;
    const int*   dst  = edge + N_EDGES;
    const int*   batch = (const int*)d_in[2];
    const float* W1   = (const float*)d_in[3];
    const float* b1   = (const float*)d_in[4];
    const float* W2   = (const float*)d_in[5];
    const float* b2   = (const float*)d_in[6];
    const float* Wout = (const float*)d_in[7];
    const float* bout = (const float*)d_in[8];
    float* out = (float*)d_out;

    // workspace carve-up (~78 MB)
    char* ws = (char*)d_ws;
    float* dinv = (float*)ws; ws += (size_t)N_NODES * sizeof(float);
    float* xw   = (float*)ws; ws += (size_t)N_NODES * CH * sizeof(float);
    float* agg  = (float*)ws; ws += (size_t)N_NODES * CH * sizeof(float);
    float* h    = (float*)ws; ws += (size_t)N_NODES * CH * sizeof(float);
    float* sums = (float*)ws; ws += (size_t)NUM_GRAPHS * CH * sizeof(float);
    float* cnt  = (float*)ws; ws += (size_t)NUM_GRAPHS * sizeof(float);

    const int ntiles = N_NODES / 16;                 // 6250
    const int gemm_blocks = (ntiles + 7) / 8;        // 8 waves per block
    const int edge_blocks = (N_EDGES * 32 + 255) / 256;
    const int node_blocks = (N_NODES * 32 + 255) / 256;
    const int comb_blocks = (N_NODES * CH + 255) / 256;

    // normalization
    hipMemsetAsync(dinv, 0, (size_t)N_NODES * sizeof(float), stream);
    k_deg <<<(N_EDGES + 255) / 256, 256, 0, stream>>>(dst, dinv);
    k_dinv<<<(N_NODES + 255) / 256, 256, 0, stream>>>(dinv);

    // layer 1
    k_gemm64<<<gemm_blocks, 256, 0, stream>>>(x, W1, xw, N_NODES);
    hipMemsetAsync(agg, 0, (size_t)N_NODES * CH * sizeof(float), stream);
    k_scatter<<<edge_blocks, 256, 0, stream>>>(src, dst, dinv, xw, agg);
    k_combine<<<comb_blocks, 256, 0, stream>>>(agg, xw, dinv, b1, h);

    // layer 2 (reuse xw/agg buffers; h overwritten in place by combine)
    k_gemm64<<<gemm_blocks, 256, 0, stream>>>(h, W2, xw, N_NODES);
    hipMemsetAsync(agg, 0, (size_t)N_NODES * CH * sizeof(float), stream);
    k_scatter<<<edge_blocks, 256, 0, stream>>>(src, dst, dinv, xw, agg);
    k_combine<<<comb_blocks, 256, 0, stream>>>(agg, xw, dinv, b2, h);

    // pool + readout (sums and cnt are contiguous -> one memset)
    hipMemsetAsync(sums, 0, (size_t)(NUM_GRAPHS * CH + NUM_GRAPHS) * sizeof(float), stream);
    k_pool<<<node_blocks, 256, 0, stream>>>(batch, h, sums, cnt);
    k_out <<<(NUM_GRAPHS * 32 + 255) / 256, 256, 0, stream>>>(sums, cnt, Wout, bout, out);
}